// Warp_41446434406635
// MI455X (gfx1250) — compile-verified
//
#include <hip/hip_runtime.h>

// Bilinear warp, NHWC, N=4 H=1024 W=1024 C=8, zero padding outside image.
// Memory-bound op (~288 MB @ 23.3 TB/s ~ 12 us floor). Strategy:
//  - b128 vector loads/stores (8 ch = 2 x float4 per pixel)
//  - non-temporal output stores (streamed, keep L2 for the gather tiles)
//  - flow vector staged via gfx1250 async global->LDS copy (ASYNCcnt path)

typedef __attribute__((ext_vector_type(4))) float v4f;
typedef __attribute__((ext_vector_type(2))) int   v2i;

#if defined(__gfx1250__) && __has_builtin(__builtin_amdgcn_global_load_async_to_lds_b64)
#define WARP_USE_ASYNC_LDS 1
#else
#define WARP_USE_ASYNC_LDS 0
#endif

__global__ __launch_bounds__(256) void warp_bilinear_kernel(
    const float* __restrict__ x,   // [N,H,W,8]
    const float* __restrict__ u,   // [N,H,W,2]
    float* __restrict__ out)       // [N,H,W,8]
{
    constexpr int W = 1024;
    constexpr int H = 1024;
    const int tid = threadIdx.x;
    const int p   = blockIdx.x * 256 + tid;   // linear pixel id over N*H*W

    float dx, dy;
#if WARP_USE_ASYNC_LDS
    __shared__ float sflow[256 * 2];
    {
        // Builtin wants: (v2i addrspace(1)* src, v2i addrspace(3)* dst, offset, cpol)
        __attribute__((address_space(1))) v2i* gsrc =
            (__attribute__((address_space(1))) v2i*)(u + (size_t)p * 2);
        __attribute__((address_space(3))) v2i* ldst =
            (__attribute__((address_space(3))) v2i*)(&sflow[tid * 2]);
        // global_load_async_to_lds_b64: per-lane 8B flow vector -> LDS
        __builtin_amdgcn_global_load_async_to_lds_b64(gsrc, ldst, 0, 0);
    }
#if __has_builtin(__builtin_amdgcn_s_wait_asynccnt)
    __builtin_amdgcn_s_wait_asynccnt(0);
#else
    asm volatile("s_wait_asynccnt 0" ::: "memory");
#endif
    __syncthreads();
    dx = sflow[tid * 2 + 0];
    dy = sflow[tid * 2 + 1];
#else
    const float2 fl = ((const float2*)u)[p];
    dx = fl.x;
    dy = fl.y;
#endif

    // Decode pixel coordinates (W, H are powers of two).
    const int xi    = p & (W - 1);
    const int yi    = (p >> 10) & (H - 1);
    const int nbase = p & ~((H * W) - 1);    // n * H * W

    const float gx  = (float)xi + dx;
    const float gy  = (float)yi + dy;
    const float x0f = floorf(gx);
    const float y0f = floorf(gy);
    const float wx1 = gx - x0f;
    const float wy1 = gy - y0f;
    const float wx0 = 1.0f - wx1;
    const float wy0 = 1.0f - wy1;
    const int   cx0 = (int)x0f;
    const int   cy0 = (int)y0f;

    const v4f* __restrict__ img = (const v4f*)x;   // pixel = 2 consecutive v4f

    v4f acc0 = 0.0f;
    v4f acc1 = 0.0f;

    auto corner = [&](int yy, int xx, float w) {
        const bool valid = ((unsigned)yy < (unsigned)H) & ((unsigned)xx < (unsigned)W);
        const float ww = valid ? w : 0.0f;
        // Clamp so the (zero-weighted) load stays in-bounds.
        const int yc = min(max(yy, 0), H - 1);
        const int xc = min(max(xx, 0), W - 1);
        const int base = (nbase + yc * W + xc) * 2;    // v4f index
        const v4f v0 = img[base];       // global_load_b128
        const v4f v1 = img[base + 1];   // global_load_b128
        acc0 += v0 * ww;
        acc1 += v1 * ww;
    };

    corner(cy0,     cx0,     wy0 * wx0);
    corner(cy0,     cx0 + 1, wy0 * wx1);
    corner(cy0 + 1, cx0,     wy1 * wx0);
    corner(cy0 + 1, cx0 + 1, wy1 * wx1);

    // Streamed output: non-temporal b128 stores (don't pollute L2/WGP$,
    // which we want holding the gather tiles of x).
    v4f* o = (v4f*)out;
    __builtin_nontemporal_store(acc0, o + (size_t)p * 2);
    __builtin_nontemporal_store(acc1, o + (size_t)p * 2 + 1);
}

extern "C" void kernel_launch(void* const* d_in, const int* in_sizes, int n_in,
                              void* d_out, int out_size, void* d_ws, size_t ws_size,
                              hipStream_t stream) {
    const float* x   = (const float*)d_in[0];   // [4,1024,1024,8] f32
    const float* u   = (const float*)d_in[1];   // [4,1024,1024,2] f32
    float*       out = (float*)d_out;

    const int npix   = in_sizes[0] / 8;         // N*H*W = 4,194,304
    const int blocks = (npix + 255) / 256;      // exact multiple (16384)

    warp_bilinear_kernel<<<blocks, 256, 0, stream>>>(x, u, out);
}